// LightGCN_Encoder_66657892434464
// MI455X (gfx1250) — compile-verified
//
#include <hip/hip_runtime.h>
#include <hip/hip_bf16.h>
#include <stdint.h>

#ifndef __has_builtin
#define __has_builtin(x) 0
#endif

// ---------------------------------------------------------------------------
// LightGCN propagation on gfx1250 (MI455X):
//   3 x scatter SpMM (native no-return f32 L2 atomics) over a 100k-node /
//   2M-edge graph, EMB_DIM=64. Edge tiles are staged into LDS by the Tensor
//   Data Mover (tensor_load_to_lds + s_wait_tensorcnt), then dropout-kept
//   edges are compacted in LDS so the 16-lane-per-edge scatter loop runs at
//   full occupancy. Embedding table (25.6MB) is L2-resident (192MB L2).
// ---------------------------------------------------------------------------

#define EMB 64
#define TILE 1024          // edges per block tile
#define BLOCK 256          // 8 wave32 per block

#if defined(__HIP_DEVICE_COMPILE__) && __has_builtin(__builtin_amdgcn_tensor_load_to_lds)
#define HAVE_TDM 1
#else
#define HAVE_TDM 0
#endif

#if HAVE_TDM
typedef unsigned int u32x4 __attribute__((ext_vector_type(4)));
typedef int          i32x4 __attribute__((ext_vector_type(4)));
typedef int          i32x8 __attribute__((ext_vector_type(8)));

__device__ __forceinline__ uint32_t lds_off(const void* p) {
  // generic -> LDS addrspace cast yields the 32-bit LDS byte offset
  return (uint32_t)(size_t)(__attribute__((address_space(3))) const void*)p;
}

// DMA a contiguous 1-D run of `nelem` elements (dsize_code: 0=1B, 2=4B)
// from global memory into LDS at byte offset `lds_byte`.
__device__ __forceinline__ void tdm_copy_1d(uint32_t lds_byte, const void* gptr,
                                            uint32_t nelem, uint32_t dsize_code) {
  uint64_t ga = (uint64_t)(size_t)gptr;
  u32x4 g0;
  g0.x = 1u;                                   // count=1, is_restore=0, no gather
  g0.y = lds_byte;                             // lds_addr
  g0.z = (uint32_t)ga;                         // global_addr[31:0]
  g0.w = ((uint32_t)(ga >> 32) & 0x01FFFFFFu)  // global_addr[56:32]
         | (2u << 30);                         // type = 2 ("image")
  i32x8 g1;
  g1[0] = (int)(dsize_code << 16);             // wg_mask=0, data_size
  g1[1] = (int)((nelem & 0xFFFFu) << 16);      // tensor_dim0[15:0]
  g1[2] = (int)((nelem >> 16) | (1u << 16));   // tensor_dim0[31:16], tensor_dim1=1
  g1[3] = (int)((nelem & 0xFFFFu) << 16);      // tile_dim0 = nelem
  g1[4] = 1;                                   // tile_dim1 = 1, tile_dim2 = 0
  g1[5] = (int)nelem;                          // tensor_dim0_stride[31:0]
  g1[6] = 0;
  g1[7] = 0;
  i32x4 z4 = {0, 0, 0, 0};
#if defined(__clang_major__) && __clang_major__ >= 23
  i32x8 z8 = {0, 0, 0, 0, 0, 0, 0, 0};
  __builtin_amdgcn_tensor_load_to_lds(g0, g1, z4, z4, z8, 0);
#else
  __builtin_amdgcn_tensor_load_to_lds(g0, g1, z4, z4, 0);
#endif
}
#endif // HAVE_TDM

__device__ __forceinline__ void atomic_add_f32(float* p, float v) {
#if __has_builtin(__hip_atomic_fetch_add)
  // relaxed / agent scope, result unused -> no-return global_atomic_add_f32
  __hip_atomic_fetch_add(p, v, __ATOMIC_RELAXED, __HIP_MEMORY_SCOPE_AGENT);
#else
  atomicAdd(p, v);
#endif
}

// ---------------------------------------------------------------------------
// Edge-parallel scatter SpMM: nxt[row] += (keep ? 2*val : 0) * cur[col]
//  phase 1: TDM stages {row,col,val,mask} tile into LDS
//  phase 2: compact kept edges in LDS (removes 50% dropout divergence)
//  phase 3: 16 lanes per edge (float4 each), LDS-broadcast metadata,
//           b128 gather from cur, 4x no-return f32 atomics into nxt
// ---------------------------------------------------------------------------
__global__ void __launch_bounds__(BLOCK)
spmm_edge_kernel(const float* __restrict__ cur, float* __restrict__ nxt,
                 const int* __restrict__ row, const int* __restrict__ col,
                 const float* __restrict__ vals,
                 const unsigned char* __restrict__ keep, int n_edges) {
  __shared__ int           s_row[TILE];
  __shared__ int           s_col[TILE];
  __shared__ float         s_val[TILE];
  __shared__ unsigned char s_msk[TILE];
  __shared__ int           s_crow[TILE];
  __shared__ int           s_ccol[TILE];
  __shared__ float         s_cval[TILE];
  __shared__ int           s_cnt;

  const int base = blockIdx.x * TILE;
  int tile_n     = n_edges - base;
  if (tile_n > TILE) tile_n = TILE;

  if (threadIdx.x == 0) s_cnt = 0;

#if HAVE_TDM
  {
    // one TDM descriptor per array, issued by waves 0..3 (scalar operands)
    const int w = threadIdx.x >> 5;
    if (w == 0)      tdm_copy_1d(lds_off(s_row), row  + base, (uint32_t)tile_n, 2u);
    else if (w == 1) tdm_copy_1d(lds_off(s_col), col  + base, (uint32_t)tile_n, 2u);
    else if (w == 2) tdm_copy_1d(lds_off(s_val), vals + base, (uint32_t)tile_n, 2u);
    else if (w == 3) tdm_copy_1d(lds_off(s_msk), keep + base, (uint32_t)tile_n, 0u);
    __builtin_amdgcn_s_wait_tensorcnt(0);
  }
#else
  for (int i = threadIdx.x; i < tile_n; i += BLOCK) {
    s_row[i] = row[base + i];
    s_col[i] = col[base + i];
    s_val[i] = vals[base + i];
    s_msk[i] = keep[base + i];
  }
#endif
  __syncthreads();

  // compact kept edges (LDS ticket counter), pre-scaling by 1/(1-RATE)=2
  for (int i = threadIdx.x; i < tile_n; i += BLOCK) {
    if (s_msk[i]) {
      const int p = atomicAdd(&s_cnt, 1);
      s_crow[p] = s_row[i];
      s_ccol[p] = s_col[i];
      s_cval[p] = s_val[i] * 2.0f;
    }
  }
  __syncthreads();
  const int cnt = s_cnt;

  const int lane = threadIdx.x & 15;   // dim group within edge
  const int d0   = lane << 2;          // float4 dim offset
  for (int e = threadIdx.x >> 4; e < cnt; e += (BLOCK / 16)) {
    const float w = s_cval[e];
    const int r = s_crow[e];
    const int c = s_ccol[e];
    const float4 x = *reinterpret_cast<const float4*>(cur + (size_t)c * EMB + d0);
    float* dst = nxt + (size_t)r * EMB + d0;
    atomic_add_f32(dst + 0, w * x.x);
    atomic_add_f32(dst + 1, w * x.y);
    atomic_add_f32(dst + 2, w * x.z);
    atomic_add_f32(dst + 3, w * x.w);
  }
}

// ---------------------------------------------------------------------------
// bufA = concat(user_emb, item_emb), vectorized float4
// ---------------------------------------------------------------------------
__global__ void __launch_bounds__(BLOCK)
init_kernel(const float* __restrict__ ue, const float* __restrict__ ie,
            float* __restrict__ buf, int n_user_f, int n_total_f) {
  const int i4 = blockIdx.x * BLOCK + threadIdx.x;
  const int fi = i4 << 2;
  if (fi >= n_total_f) return;
  float4 v;
  if (fi < n_user_f) v = *reinterpret_cast<const float4*>(ue + fi);
  else               v = *reinterpret_cast<const float4*>(ie + (fi - n_user_f));
  *reinterpret_cast<float4*>(buf + fi) = v;
}

// ---------------------------------------------------------------------------
// Gather 0.25*src rows for users (out[0:nq*64]) and items (out[nq*64:]),
// accumulating into d_out so no full-size `acc` buffer is needed.
// ---------------------------------------------------------------------------
__global__ void __launch_bounds__(BLOCK)
gather_kernel(const float* __restrict__ src, const int* __restrict__ users,
              const int* __restrict__ items, float* __restrict__ out,
              int n_q, int n_users, int overwrite) {
  const int t    = blockIdx.x * BLOCK + threadIdx.x;
  const int slot = t >> 4;
  const int lane = t & 15;
  if (slot >= 2 * n_q) return;
  int node;
  if (slot < n_q) node = users[slot];
  else            node = items[slot - n_q] + n_users;
  const float4 x = *reinterpret_cast<const float4*>(src + (size_t)node * EMB + (lane << 2));
  float4 r;
  r.x = 0.25f * x.x; r.y = 0.25f * x.y; r.z = 0.25f * x.z; r.w = 0.25f * x.w;
  float4* o = reinterpret_cast<float4*>(out + (size_t)slot * EMB + (lane << 2));
  if (!overwrite) {
    const float4 p = *o;
    r.x += p.x; r.y += p.y; r.z += p.z; r.w += p.w;
  }
  *o = r;
}

// ---------------------------------------------------------------------------
extern "C" void kernel_launch(void* const* d_in, const int* in_sizes, int n_in,
                              void* d_out, int out_size, void* d_ws, size_t ws_size,
                              hipStream_t stream) {
  const float*         user_emb = (const float*)d_in[0];
  const float*         item_emb = (const float*)d_in[1];
  const float*         vals     = (const float*)d_in[2];
  const int*           row      = (const int*)d_in[3];
  const int*           col      = (const int*)d_in[4];
  const unsigned char* keep     = (const unsigned char*)d_in[5]; // jnp bool_ = 1 byte
  const int*           users    = (const int*)d_in[6];
  const int*           items    = (const int*)d_in[7];
  float*               out      = (float*)d_out;

  const int n_user_f  = in_sizes[0];
  const int n_item_f  = in_sizes[1];
  const int n_edges   = in_sizes[2];
  const int n_q       = in_sizes[6];
  const int n_users   = n_user_f / EMB;
  const int n_total_f = n_user_f + n_item_f;

  const size_t buf_bytes = (((size_t)n_total_f * sizeof(float)) + 255u) & ~(size_t)255u;
  float* bufA = (float*)d_ws;
  float* bufB = (float*)((char*)d_ws + buf_bytes);

  const int init_blocks = (n_total_f / 4 + BLOCK - 1) / BLOCK;
  const int gath_blocks = (2 * n_q * 16 + BLOCK - 1) / BLOCK;
  const int edge_blocks = (n_edges + TILE - 1) / TILE;

  // layer 0 contribution (the raw embeddings)
  init_kernel<<<init_blocks, BLOCK, 0, stream>>>(user_emb, item_emb, bufA,
                                                 n_user_f, n_total_f);
  gather_kernel<<<gath_blocks, BLOCK, 0, stream>>>(bufA, users, items, out,
                                                   n_q, n_users, /*overwrite=*/1);

  float* cur = bufA;
  float* nxt = bufB;
  for (int l = 0; l < 3; ++l) {
    hipMemsetAsync(nxt, 0, (size_t)n_total_f * sizeof(float), stream);
    spmm_edge_kernel<<<edge_blocks, BLOCK, 0, stream>>>(cur, nxt, row, col,
                                                        vals, keep, n_edges);
    gather_kernel<<<gath_blocks, BLOCK, 0, stream>>>(nxt, users, items, out,
                                                     n_q, n_users, /*overwrite=*/0);
    float* t = cur; cur = nxt; nxt = t;
  }
}